// VolIntegratedGRU_67619965108301
// MI455X (gfx1250) — compile-verified
//
#include <hip/hip_runtime.h>
#include <stdint.h>

// ---------------------------------------------------------------------------
// VolIntegratedGRU on MI455X (gfx1250):
//   - serial nonlinear scan  -> single wave32, lane0 recurrence
//   - critical chain on HW transcendentals; gfx1250 v_tanh_f32 if available
//     (sigmoid via 0.5+0.5*tanh(x/2): 1 trans op instead of exp2+rcp)
//   - inputs double-buffered into LDS via global_load_async_to_lds_b128
//     (ASYNCcnt), one s_wait_asynccnt per 1024-step chunk
//   - var(y) seed computed by a parallel reduction kernel (atomic f32 adds)
// ---------------------------------------------------------------------------

#define T_LEN   262144
#define CHUNK   1024
#define NSTEPS  (T_LEN - 1)
#define NCHUNK  ((NSTEPS + CHUNK - 1) / CHUNK)   // 256
#define YBUF_F4 288                               // 9 reps * 32 lanes (float4 units)
#define RVBUF_F4 256                              // 8 reps * 32 lanes

// ----- fast device math on CDNA5 hardware transcendentals -------------------
__device__ __forceinline__ float fexp2(float x) { return __builtin_amdgcn_exp2f(x); }
__device__ __forceinline__ float flog2(float x) { return __builtin_amdgcn_logf(x); }   // v_log_f32 = log2
__device__ __forceinline__ float frcp (float x) { return __builtin_amdgcn_rcpf(x); }
__device__ __forceinline__ float frsq (float x) { return __builtin_amdgcn_rsqf(x); }
__device__ __forceinline__ float fsqrtf_(float x){ return __builtin_amdgcn_sqrtf(x); }

#define LOG2E 1.4426950408889634f
#define LN2   0.6931471805599453f

// gfx1250 adds a hardware v_tanh_f32 transcendental; use it if the toolchain
// exposes a builtin, otherwise fall back to exp2/rcp.
#if __has_builtin(__builtin_amdgcn_tanhf)
__device__ __forceinline__ float tanh_hw(float x) { return __builtin_amdgcn_tanhf(x); }
#define HAS_HW_TANH 1
#elif __has_builtin(__builtin_amdgcn_tanh_f32)
__device__ __forceinline__ float tanh_hw(float x) { return __builtin_amdgcn_tanh_f32(x); }
#define HAS_HW_TANH 1
#else
#define HAS_HW_TANH 0
#endif

__device__ __forceinline__ float tanh_f(float x) {
#if HAS_HW_TANH
    return tanh_hw(x);                       // 1 dependent trans op
#else
    // tanh(x) = 1 - 2/(exp(2x)+1)           // 2 dependent trans ops
    return 1.0f - 2.0f * frcp(fexp2(2.0f * LOG2E * x) + 1.0f);
#endif
}
__device__ __forceinline__ float sigmoid_f(float x) {
#if HAS_HW_TANH
    // sigmoid(x) = 0.5 + 0.5*tanh(x/2)      // 1 dependent trans op
    return __builtin_fmaf(0.5f, tanh_hw(0.5f * x), 0.5f);
#else
    return frcp(1.0f + fexp2(-LOG2E * x));
#endif
}
__device__ __forceinline__ float softplus_f(float x) {
    float sp = LN2 * flog2(1.0f + fexp2(LOG2E * x));
    return (x > 30.0f) ? x : sp;   // avoid exp2 overflow for large x
}

// ----- async copy: global float4s -> LDS, all 32 lanes cooperate ------------
// gbase: uniform 64-bit base (SGPR pair), per-lane 32-bit byte offset (VGPR),
// per-lane LDS byte address (VGPR).  Tracked by ASYNCcnt.
__device__ __forceinline__ void async_copy_f4(const float* gbase,
                                              uint32_t g_f4_off,
                                              uint32_t g_f4_limit,
                                              float* lds_dst,
                                              int reps) {
    const uint32_t lane   = threadIdx.x & 31u;
    const uint32_t ldsoff = (uint32_t)(uintptr_t)lds_dst;  // low 32 bits of generic ptr = LDS offset
    for (int k = 0; k < reps; ++k) {
        uint32_t f4  = (uint32_t)(k * 32) + lane;
        uint32_t gf4 = g_f4_off + f4;
        if (gf4 < g_f4_limit) {                      // EXEC-masked tail guard
            uint32_t ldsb = ldsoff + f4 * 16u;
            uint32_t gb   = gf4 * 16u;
            asm volatile("global_load_async_to_lds_b128 %0, %1, %2"
                         :
                         : "v"(ldsb), "v"(gb), "s"(gbase)
                         : "memory");
        }
    }
}

__device__ __forceinline__ void wait_async0() {
    asm volatile("s_wait_asynccnt 0x0" ::: "memory");
}

// ----- workspace init --------------------------------------------------------
__global__ void init_acc_kernel(float* acc) {
    acc[0] = 0.0f;   // sum(y)
    acc[1] = 0.0f;   // sum(y^2)
}

// ----- parallel reduction for var(y) ----------------------------------------
__global__ void reduce_var_kernel(const float* __restrict__ y, int n, float* acc) {
    __shared__ float s1[256];
    __shared__ float s2[256];
    float a = 0.0f, b = 0.0f;
    for (int i = blockIdx.x * blockDim.x + threadIdx.x; i < n;
         i += gridDim.x * blockDim.x) {
        float v = y[i];
        a += v;
        b += v * v;
    }
    s1[threadIdx.x] = a;
    s2[threadIdx.x] = b;
    __syncthreads();
    for (int s = 128; s > 0; s >>= 1) {
        if (threadIdx.x < s) {
            s1[threadIdx.x] += s1[threadIdx.x + s];
            s2[threadIdx.x] += s2[threadIdx.x + s];
        }
        __syncthreads();
    }
    if (threadIdx.x == 0) {
        atomicAdd(&acc[0], s1[0]);
        atomicAdd(&acc[1], s2[0]);
    }
}

// ----- the serial scan: one wave32 ------------------------------------------
__global__ __launch_bounds__(32, 1)
void scan_kernel(const float* __restrict__ y, const float* __restrict__ rv,
                 const float* omega_p, const float* alpha_p, const float* phi_p,
                 const float* lam_p, const float* gam_p, const float* gamma_u_p,
                 const float* W_r, const float* U_r, const float* b_r,
                 const float* W_u, const float* U_u, const float* b_u,
                 const float* W_n, const float* U_n, const float* b_n,
                 const float* acc, float* __restrict__ out) {
    // double-buffered input staging in LDS
    __shared__ float ybuf [2][YBUF_F4 * 4];   // covers y[base .. base+1151]
    __shared__ float rvbuf[2][RVBUF_F4 * 4];  // covers rv[base .. base+1023]

    const int lane = threadIdx.x;

    // ---- hoisted scalars (uniform SGPR/VGPR constants) ----
    const float omega = omega_p[0], alpha = alpha_p[0], phi = phi_p[0];
    const float lam = lam_p[0], gam = gam_p[0], gu = gamma_u_p[0];
    const float wr1 = W_r[1], wr2 = W_r[2], wr3 = W_r[3];
    const float wu1 = W_u[1], wu2 = W_u[2], wu3 = W_u[3];
    const float wn1 = W_n[1], wn2 = W_n[2], wn3 = W_n[3];
    const float cr = W_r[0] + U_r[0];   // fold eta*W[0] + eta*U into one FMA
    const float cu = W_u[0] + U_u[0];
    const float wn0 = W_n[0], un = U_n[0];
    const float br = b_r[0], bu = b_u[0], bn = b_n[0];

    // seed from the reduction
    const float inv_t = 1.0f / (float)T_LEN;
    const float mean  = acc[0] * inv_t;
    const float hvol0 = acc[1] * inv_t - mean * mean;  // population var

    float eta  = hvol0;
    float hvol = hvol0;
    float z    = (y[0] - lam * hvol0) * frsq(hvol0);

    if (lane == 0) {
        out[0]     = z;       // z_val[0]  = z0
        out[T_LEN] = hvol0;   // h_val[0]  = hvol0
    }

    const uint32_t ylim  = T_LEN / 4;   // float4 count of y
    const uint32_t rvlim = T_LEN / 4;

    // prologue: prefetch chunk 0
    async_copy_f4(y,  0u, ylim,  &ybuf [0][0], 9);
    async_copy_f4(rv, 0u, rvlim, &rvbuf[0][0], 8);

    for (int c = 0; c < NCHUNK; ++c) {
        wait_async0();                          // chunk c resident in LDS

        if (c + 1 < NCHUNK) {                   // fire-and-forget prefetch of c+1
            uint32_t nbase4 = (uint32_t)((c + 1) * (CHUNK / 4));
            int nb = (c + 1) & 1;
            async_copy_f4(y,  nbase4, ylim,  &ybuf [nb][0], 9);
            async_copy_f4(rv, nbase4, rvlim, &rvbuf[nb][0], 8);
        }

        const int base  = c * CHUNK;
        const int steps = (NSTEPS - base) < CHUNK ? (NSTEPS - base) : CHUNK;
        const int b     = c & 1;

        if (lane == 0) {
            for (int i = 0; i < steps; ++i) {
                const float y_prev  = ybuf [b][i];
                const float rv_prev = rvbuf[b][i];
                const float y_cur   = ybuf [b][i + 1];

                // gate pre-activations; y/rv terms are off the dependent chain
                const float ar = br + y_prev * wr1 + rv_prev * wr2
                               + eta * cr + z * wr3;
                const float au = bu + y_prev * wu1 + rv_prev * wu2 + gu * hvol
                               + eta * cu + z * wu3;
                const float r  = sigmoid_f(ar);
                const float u  = sigmoid_f(au);
                const float an = bn + y_prev * wn1 + rv_prev * wn2
                               + eta * wn0 + z * wn3 + r * eta * un;
                const float n  = tanh_f(an);

                const float h       = (1.0f - u) * n + u * eta;
                const float eta_new = softplus_f(h);

                // hvol chain runs in parallel with the eta chain (ILP)
                const float hv_new = omega + phi * (hvol - omega)
                    + alpha * (z * z - 1.0f - 2.0f * gam * fsqrtf_(hvol) * z);

                const float z_new = (y_cur - lam * eta_new) * frsq(eta_new);

                const int t = base + i;
                out[1 + t]         = z_new;   // fire-and-forget stores
                out[T_LEN + 1 + t] = hv_new;

                eta = eta_new; hvol = hv_new; z = z_new;
            }
        }
    }
    wait_async0();   // drain before endpgm
}

// ---------------------------------------------------------------------------
extern "C" void kernel_launch(void* const* d_in, const int* in_sizes, int n_in,
                              void* d_out, int out_size, void* d_ws, size_t ws_size,
                              hipStream_t stream) {
    const float* y        = (const float*)d_in[0];
    const float* rv       = (const float*)d_in[1];
    const float* omega_p  = (const float*)d_in[2];
    const float* alpha_p  = (const float*)d_in[3];
    const float* phi_p    = (const float*)d_in[4];
    const float* lam_p    = (const float*)d_in[5];
    const float* gam_p    = (const float*)d_in[6];
    const float* gamma_up = (const float*)d_in[7];
    const float* W_r      = (const float*)d_in[8];
    const float* U_r      = (const float*)d_in[9];
    const float* b_r      = (const float*)d_in[10];
    const float* W_u      = (const float*)d_in[11];
    const float* U_u      = (const float*)d_in[12];
    const float* b_u      = (const float*)d_in[13];
    const float* W_n      = (const float*)d_in[14];
    const float* U_n      = (const float*)d_in[15];
    const float* b_n      = (const float*)d_in[16];
    float* out = (float*)d_out;
    float* acc = (float*)d_ws;   // 2 floats: sum, sumsq

    init_acc_kernel<<<1, 1, 0, stream>>>(acc);
    reduce_var_kernel<<<256, 256, 0, stream>>>(y, T_LEN, acc);
    scan_kernel<<<1, 32, 0, stream>>>(y, rv,
                                      omega_p, alpha_p, phi_p, lam_p, gam_p, gamma_up,
                                      W_r, U_r, b_r, W_u, U_u, b_u, W_n, U_n, b_n,
                                      acc, out);
}